// DetectionPostProcess_15719580304029
// MI455X (gfx1250) — compile-verified
//
#include <hip/hip_runtime.h>

#define NTOT  65472
#define NIMG  4
#define NCLS  80
#define PRE   1000
#define POST  300
#define NT    1024

// CDNA5 async global->LDS DMA via inline asm (builtin prototype uses CUDA
// language address spaces that generic HIP pointers can't convert to).
#if defined(__gfx1250__)
#define USE_ASYNC_LDS 1
#endif

// Monotone float -> sortable uint key (larger key == larger float)
__device__ __forceinline__ unsigned f2k(float f) {
  unsigned u = __float_as_uint(f);
  return (u & 0x80000000u) ? ~u : (u | 0x80000000u);
}
__device__ __forceinline__ float k2f(unsigned k) {
  unsigned u = (k & 0x80000000u) ? (k ^ 0x80000000u) : ~k;
  return __uint_as_float(u);
}

// ---------------- K1: per-level decode (bandwidth-bound part) -------------
__global__ __launch_bounds__(256) void decode_kernel(
    const float* __restrict__ boxes, const float* __restrict__ scores,
    const float* __restrict__ regr, float* __restrict__ dbox,
    float* __restrict__ dsc, int* __restrict__ dlab,
    int n, int goff, int total) {
  int t = blockIdx.x * blockDim.x + threadIdx.x;
  if (t >= total) return;

  // warm the small box/regr rows (near scope) while the score row streams
  __builtin_prefetch(boxes + (size_t)t * 4, 0, 3);
  __builtin_prefetch(regr + (size_t)t * 4, 0, 3);

  const float4* s4 = (const float4*)(scores + (size_t)t * NCLS);

  // First vector: consume q0.x (background class) in a way the compiler
  // cannot fold away, so the b128 loads stay 16B-aligned (offset 0,16,...).
  // For any finite q0.x this initializes best = -inf, matching s[0] = -inf.
  float4 q0 = s4[0];
  float best = (q0.x == q0.x) ? -__builtin_inff() : q0.x;
  int lab = 0;
  if (q0.y > best) { best = q0.y; lab = 1; }
  if (q0.z > best) { best = q0.z; lab = 2; }
  if (q0.w > best) { best = q0.w; lab = 3; }
#pragma unroll
  for (int v = 1; v < NCLS / 4; ++v) {
    float4 q = s4[v];
    int base = 4 * v;
    if (q.x > best) { best = q.x; lab = base; }
    if (q.y > best) { best = q.y; lab = base + 1; }
    if (q.z > best) { best = q.z; lab = base + 2; }
    if (q.w > best) { best = q.w; lab = base + 3; }
  }

  float4 bb = ((const float4*)boxes)[t];
  float4 rr = ((const float4*)regr)[t];
  float w = bb.z - bb.x, h = bb.w - bb.y;
  float cx = bb.x + 0.5f * w, cy = bb.y + 0.5f * h;
  float pcx = rr.x * 0.1f * w + cx;
  float pcy = rr.y * 0.1f * h + cy;
  float pw = expf(rr.z * 0.2f) * w;
  float ph = expf(rr.w * 0.2f) * h;
  float x1 = fminf(fmaxf(pcx - 0.5f * pw, 0.f), 1024.f);
  float y1 = fminf(fmaxf(pcy - 0.5f * ph, 0.f), 1024.f);
  float x2 = fminf(fmaxf(pcx + 0.5f * pw, 0.f), 1024.f);
  float y2 = fminf(fmaxf(pcy + 0.5f * ph, 0.f), 1024.f);
  bool valid = (best > 0.05f) && ((x2 - x1) >= 2.0f) && ((y2 - y1) >= 2.0f);

  int b = t / n, a = t - b * n;
  size_t o = (size_t)b * NTOT + goff + a;
  ((float4*)dbox)[o] = make_float4(x1, y1, x2, y2);
  dsc[o] = valid ? best : -1.0f;
  dlab[o] = lab;
}

// -------- K2: per-image exact top-1000 + sort + greedy NMS + emit ---------
__global__ __launch_bounds__(NT) void select_nms_kernel(
    const float* __restrict__ dbox, const float* __restrict__ dsc,
    const int* __restrict__ dlab, float* __restrict__ fb,
    float* __restrict__ fs, float* __restrict__ fl) {
  __shared__ unsigned s_key[NT];
  __shared__ unsigned s_idx[NT];
  __shared__ float s_bx[PRE * 4];
  __shared__ unsigned s_keep[PRE];
  __shared__ int s_scan[NT];
  __shared__ int s_red[32];
  __shared__ unsigned s_cnt;

  const int tid = threadIdx.x;
  const int img = blockIdx.x;
  const float* sc = dsc + (size_t)img * NTOT;

  // --- exact 1000th-largest key: 32-step bitwise binary search over keys ---
  // (scores are L2-resident: 256KB/image; re-scan is nearly free)
  unsigned T = 0u;
  for (int bit = 31; bit >= 0; --bit) {
    unsigned cand = T | (1u << bit);
    int c = 0;
    for (int i = tid; i < NTOT; i += NT) c += (f2k(sc[i]) >= cand) ? 1 : 0;
    for (int o = 16; o > 0; o >>= 1) c += __shfl_down(c, o, 32);  // wave32
    if ((tid & 31) == 0) s_red[tid >> 5] = c;
    __syncthreads();
    if (tid < 32) {
      int x = s_red[tid];
      for (int o = 16; o > 0; o >>= 1) x += __shfl_down(x, o, 32);
      if (tid == 0) s_red[0] = x;
    }
    __syncthreads();
    if (s_red[0] >= PRE) T = cand;  // uniform across block
    __syncthreads();
  }

  // --- compact: strictly-greater first (<=999), then ties fill to 1000 ---
  if (tid == 0) s_cnt = 0u;
  __syncthreads();
  for (int i = tid; i < NTOT; i += NT) {
    unsigned k = f2k(sc[i]);
    if (k > T) {
      unsigned p = atomicAdd(&s_cnt, 1u);
      if (p < NT) { s_key[p] = k; s_idx[p] = (unsigned)i; }
    }
  }
  __syncthreads();
  for (int i = tid; i < NTOT; i += NT) {
    unsigned k = f2k(sc[i]);
    if (k == T) {
      unsigned p = atomicAdd(&s_cnt, 1u);
      if (p < PRE) { s_key[p] = k; s_idx[p] = (unsigned)i; }
    }
  }
  __syncthreads();
  if (tid >= PRE) { s_key[tid] = 0u; s_idx[tid] = 0xFFFFFFFFu; }
  __syncthreads();

  // --- bitonic sort 1024 (desc by key, asc index tie-break: deterministic,
  //     matches jax top_k stability) ---
  for (unsigned ksz = 2; ksz <= NT; ksz <<= 1) {
    for (unsigned j = ksz >> 1; j > 0; j >>= 1) {
      unsigned i = (unsigned)tid, p = i ^ j;
      if (p > i) {
        unsigned ka = s_key[i], kb = s_key[p];
        unsigned ia = s_idx[i], ib = s_idx[p];
        bool up = ((i & ksz) == 0);
        bool sw = up ? (ka < kb || (ka == kb && ia > ib))
                     : (ka > kb || (ka == kb && ia < ib));
        if (sw) { s_key[i] = kb; s_key[p] = ka; s_idx[i] = ib; s_idx[p] = ia; }
      }
      __syncthreads();
    }
  }

  unsigned myk = s_key[tid];
  unsigned myidx = s_idx[tid];
  float mysc = k2f(myk);
  int mylab = 0;
  if (tid < PRE) mylab = dlab[(size_t)img * NTOT + myidx];

  // --- gather top-1000 boxes into LDS: CDNA5 async DMA gather path.
  //     LDS dest offset per lane = low 32 bits of the generic shared pointer
  //     (ISA 10.2: LDS aperture addresses truncate to addr[31:0]). ---
  if (tid < PRE) {
    const float4* src = ((const float4*)dbox) + ((size_t)img * NTOT + myidx);
#ifdef USE_ASYNC_LDS
    unsigned lds_off = (unsigned)(unsigned long long)(const void*)&s_bx[4 * tid];
    unsigned long long gaddr = (unsigned long long)src;
    asm volatile("global_load_async_to_lds_b128 %0, %1, off"
                 :
                 : "v"(lds_off), "v"(gaddr)
                 : "memory");
#else
    float4 v = *src;
    s_bx[4 * tid + 0] = v.x; s_bx[4 * tid + 1] = v.y;
    s_bx[4 * tid + 2] = v.z; s_bx[4 * tid + 3] = v.w;
#endif
  }
#ifdef USE_ASYNC_LDS
  asm volatile("s_wait_asynccnt 0x0" ::: "memory");
#endif
  __syncthreads();

  // --- greedy NMS, exact sequential semantics of the reference fori_loop ---
  float bx1 = 0.f, by1 = 0.f, bx2 = 0.f, by2 = 0.f, area = 0.f;
  bool mk = false;
  if (tid < PRE) {
    bx1 = s_bx[4 * tid + 0]; by1 = s_bx[4 * tid + 1];
    bx2 = s_bx[4 * tid + 2]; by2 = s_bx[4 * tid + 3];
    area = (bx2 - bx1) * (by2 - by1);
    mk = (mysc > 0.05f);
    s_keep[tid] = mk ? 1u : 0u;
  }
  __syncthreads();
  for (int i = 0; i < PRE - 1; ++i) {
    if (s_keep[i] && mk && tid > i && tid < PRE) {
      float ox1 = s_bx[4 * i + 0], oy1 = s_bx[4 * i + 1];
      float ox2 = s_bx[4 * i + 2], oy2 = s_bx[4 * i + 3];
      float oarea = (ox2 - ox1) * (oy2 - oy1);
      float ix1 = fmaxf(bx1, ox1), iy1 = fmaxf(by1, oy1);
      float ix2 = fminf(bx2, ox2), iy2 = fminf(by2, oy2);
      float inter = fmaxf(ix2 - ix1, 0.f) * fmaxf(iy2 - iy1, 0.f);
      float uni = fmaxf(area + oarea - inter, 1e-6f);
      if (inter > 0.5f * uni) { mk = false; s_keep[tid] = 0u; }
    }
    __syncthreads();
  }

  // --- compact kept (already score-descending) to first 300, fill rest ---
  int flag = (tid < PRE && mk) ? 1 : 0;
  s_scan[tid] = flag;
  __syncthreads();
  for (int off = 1; off < NT; off <<= 1) {
    int add = (tid >= off) ? s_scan[tid - off] : 0;
    __syncthreads();
    s_scan[tid] += add;
    __syncthreads();
  }
  int pos = s_scan[tid] - flag;
  int total = s_scan[NT - 1];
  if (flag && pos < POST) {
    size_t o = (size_t)img * POST + pos;
    ((float4*)fb)[o] = make_float4(bx1, by1, bx2, by2);
    fs[o] = mysc;
    fl[o] = (float)mylab;
  }
  int start = total < POST ? total : POST;
  for (int p = start + tid; p < POST; p += NT) {
    size_t o = (size_t)img * POST + p;
    ((float4*)fb)[o] = make_float4(0.f, 0.f, 0.f, 0.f);
    fs[o] = -1.0f;
    fl[o] = -1.0f;
  }
}

extern "C" void kernel_launch(void* const* d_in, const int* in_sizes, int n_in,
                              void* d_out, int out_size, void* d_ws, size_t ws_size,
                              hipStream_t stream) {
  (void)in_sizes; (void)n_in; (void)out_size; (void)ws_size;
  static const int NS_[5] = {49152, 12288, 3072, 768, 192};
  static const int OFF_[5] = {0, 49152, 61440, 64512, 65280};

  float* dbox = (float*)d_ws;                         // [B, NTOT, 4]
  float* dsc  = dbox + (size_t)NIMG * NTOT * 4;       // [B, NTOT]
  int*   dlab = (int*)(dsc + (size_t)NIMG * NTOT);    // [B, NTOT]

  for (int l = 0; l < 5; ++l) {
    const float* b = (const float*)d_in[3 * l + 0];
    const float* s = (const float*)d_in[3 * l + 1];
    const float* r = (const float*)d_in[3 * l + 2];
    int total = NIMG * NS_[l];
    decode_kernel<<<(total + 255) / 256, 256, 0, stream>>>(
        b, s, r, dbox, dsc, dlab, NS_[l], OFF_[l], total);
  }

  float* out = (float*)d_out;
  float* fb = out;                                    // [B, 300, 4]
  float* fs = out + (size_t)NIMG * POST * 4;          // [B, 300]
  float* fl = fs + (size_t)NIMG * POST;               // [B, 300]
  select_nms_kernel<<<NIMG, NT, 0, stream>>>(dbox, dsc, dlab, fb, fs, fl);
}